// Encoder2_59287728554196
// MI455X (gfx1250) — compile-verified
//
#include <hip/hip_runtime.h>
#include <math.h>

typedef float v2f __attribute__((ext_vector_type(2)));
typedef float v8f __attribute__((ext_vector_type(8)));

#define IN_F   128
#define OUT_F  64
#define NB     8                    // NUM + K = 5 + 3
#define KZ     (IN_F + IN_F * NB)   // 1152 : silu part + spline part
#define SZ     (KZ / 4)             // 288 K-steps for the z GEMM
#define SX     (IN_F / 4)           // 32 K-steps for the x_lin GEMM
#define WZ_FLOATS (SZ * 256)        // fragment-ordered Wz in ws
#define WX_FLOATS (SX * 256)

// LDS fragment index: [step][M 0..15][k 0..3], rotated by 16B per step to kill
// bank conflicts for both the phase-1 scattered stores and the WMMA A-fetch.
__device__ __forceinline__ int fidx(int s, int m, int kl) {
    return s * 64 + (((m << 2) + ((s & 15) << 2)) & 63) + kl;
}

// ---------------- prep: build fragment-ordered weight matrices in ws --------
// Wz[r][o], r<128 : scale_base*mask ; r>=128 : coef[i,o,k]*scale_sp*mask,
//   r = 128 + i*8 + k.   Stored as ws[(r>>2)*256 + o*4 + (r&3)].
// Wx[r][o] = W1[o][r]  stored after Wz with the same fragment layout.
__global__ __launch_bounds__(256) void kan_prep(
    const float* __restrict__ W1,   const float* __restrict__ coef,
    const float* __restrict__ sb,   const float* __restrict__ sp,
    const float* __restrict__ mask, float* __restrict__ ws)
{
    int idx = blockIdx.x * blockDim.x + threadIdx.x;
    if (idx < KZ * OUT_F) {
        int r = idx >> 6, o = idx & 63;
        float v;
        if (r < IN_F) {
            v = sb[r * OUT_F + o] * mask[r * OUT_F + o];
        } else {
            int rr = r - IN_F, i = rr >> 3, k = rr & 7;
            v = coef[(i * OUT_F + o) * NB + k] * sp[i * OUT_F + o] * mask[i * OUT_F + o];
        }
        ws[(r >> 2) * 256 + o * 4 + (r & 3)] = v;
    } else if (idx < KZ * OUT_F + IN_F * OUT_F) {
        int j = idx - KZ * OUT_F;
        int r = j >> 6, o = j & 63;
        ws[WZ_FLOATS + (r >> 2) * 256 + o * 4 + (r & 3)] = W1[o * IN_F + r];
    }
}

// ---------------- fused KAN kernel: 16 rows / block, 128 threads (4 waves) --
__global__ __launch_bounds__(128) void kan_fused(
    const float* __restrict__ x, const float* __restrict__ grid,
    const float* __restrict__ wfrag, float* __restrict__ out, int N)
{
    __shared__ float ldsA[SZ * 64];   // [silu | spline] activation fragments
    __shared__ float ldsX[SX * 64];   // raw-x fragments for x_lin
    __shared__ float rowsq[16];

    const int t    = threadIdx.x;       // 0..127 ; thread t owns feature t
    const int lane = t & 31;
    const int wv   = t >> 5;            // wave id -> output column slice
    const int row0 = blockIdx.x * 16;

    if (t < 16) rowsq[t] = 0.0f;

    // ---- phase 1: per-feature grid, silu + B-spline bases into LDS --------
    float g[12];
#pragma unroll
    for (int j = 0; j < 12; ++j) g[j] = grid[t * 12 + j];

    float rd1[11], rd2[10], rd3[9];     // reciprocal knot gaps (no NaN: uniform)
#pragma unroll
    for (int j = 0; j < 11; ++j) rd1[j] = 1.0f / (g[j + 1] - g[j]);
#pragma unroll
    for (int j = 0; j < 10; ++j) rd2[j] = 1.0f / (g[j + 2] - g[j]);
#pragma unroll
    for (int j = 0; j < 9;  ++j) rd3[j] = 1.0f / (g[j + 3] - g[j]);

#pragma unroll 2
    for (int m = 0; m < 16; ++m) {
        const int r = row0 + m;
        const float xv  = (r < N) ? x[r * IN_F + t] : 0.0f;
        const float sil = xv / (1.0f + __expf(-xv));
        ldsX[fidx(t >> 2, m, t & 3)] = xv;
        ldsA[fidx(t >> 2, m, t & 3)] = sil;   // k-index = t (cols 0..127)

        float B[11];
#pragma unroll
        for (int j = 0; j < 11; ++j)
            B[j] = (xv >= g[j] && xv < g[j + 1]) ? 1.0f : 0.0f;
#pragma unroll
        for (int j = 0; j < 10; ++j) {
            float v = (xv - g[j]) * rd1[j] * B[j] + (g[j + 2] - xv) * rd1[j + 1] * B[j + 1];
            B[j] = (v == v) ? v : 0.0f;
        }
#pragma unroll
        for (int j = 0; j < 9; ++j) {
            float v = (xv - g[j]) * rd2[j] * B[j] + (g[j + 3] - xv) * rd2[j + 1] * B[j + 1];
            B[j] = (v == v) ? v : 0.0f;
        }
#pragma unroll
        for (int j = 0; j < 8; ++j) {
            float v = (xv - g[j]) * rd3[j] * B[j] + (g[j + 4] - xv) * rd3[j + 1] * B[j + 1];
            B[j] = (v == v) ? v : 0.0f;
        }
        // spline k-indices 128 + 8t .. +7  ->  steps s0, s0+1 (kl 0..3 each)
        const int s0 = (IN_F + t * 8) >> 2;
        float4 lo = make_float4(B[0], B[1], B[2], B[3]);
        float4 hi = make_float4(B[4], B[5], B[6], B[7]);
        *(float4*)&ldsA[fidx(s0,     m, 0)] = lo;
        *(float4*)&ldsA[fidx(s0 + 1, m, 0)] = hi;
    }
    __syncthreads();

    // ---- phase 2: WMMA GEMMs (f32 16x16x4) --------------------------------
    const int koff = (lane >> 4) << 1;                  // 0 | 2 (ISA A/B layout)
    const int aoff = ((lane & 15) << 2) + koff;         // M*4 + koff (pre-rotation)
    const int colq = ((lane & 15) << 2) + koff;         // col*4 + koff within 64-f qtr
    const float* wz = wfrag + wv * 64 + colq;
    const float* wx = wfrag + WZ_FLOATS + wv * 64 + colq;

    v8f acc  = {0.f, 0.f, 0.f, 0.f, 0.f, 0.f, 0.f, 0.f};
    v8f accx = {0.f, 0.f, 0.f, 0.f, 0.f, 0.f, 0.f, 0.f};

#pragma unroll 8
    for (int s = 0; s < SZ; ++s) {
        v2f a = *(const v2f*)(ldsA + fidx(s, lane & 15, koff));
        v2f b = *(const v2f*)(wz + s * 256);
        acc = __builtin_amdgcn_wmma_f32_16x16x4_f32(
                  false, a, false, b, (short)0, acc, false, false);
    }
#pragma unroll 8
    for (int s = 0; s < SX; ++s) {
        v2f a = *(const v2f*)(ldsX + fidx(s, lane & 15, koff));
        v2f b = *(const v2f*)(wx + s * 256);
        accx = __builtin_amdgcn_wmma_f32_16x16x4_f32(
                   false, a, false, b, (short)0, accx, false, false);
    }

    // ---- phase 3: outputs --------------------------------------------------
    // C/D layout: vgpr v, lane l -> row = v + 8*(l>=16), col = l&15
    const int oc    = wv * 16 + (lane & 15);
    const int rbase = (lane >> 4) << 3;
    const int xloff = N * OUT_F;                        // x_lin after z in d_out

#pragma unroll
    for (int v = 0; v < 8; ++v) {
        int rr = row0 + rbase + v;
        if (rr < N) out[xloff + rr * OUT_F + oc] = accx[v];
    }

    // row sum-of-squares: reduce over the 16 lanes sharing a row, then LDS atomics
    float s2[8];
#pragma unroll
    for (int v = 0; v < 8; ++v) s2[v] = acc[v] * acc[v];
#pragma unroll
    for (int msk = 1; msk < 16; msk <<= 1) {
#pragma unroll
        for (int v = 0; v < 8; ++v) s2[v] += __shfl_xor(s2[v], msk, 32);
    }
    if ((lane & 15) == 0) {
#pragma unroll
        for (int v = 0; v < 8; ++v) atomicAdd(&rowsq[rbase + v], s2[v]);
    }
    __syncthreads();

#pragma unroll
    for (int v = 0; v < 8; ++v) {
        int rloc = rbase + v;
        int rr   = row0 + rloc;
        float sc = 0.8f / fmaxf(sqrtf(rowsq[rloc]), 1e-12f);
        if (rr < N) out[rr * OUT_F + oc] = acc[v] * sc;
    }
}

// ---------------------------------------------------------------------------
extern "C" void kernel_launch(void* const* d_in, const int* in_sizes, int n_in,
                              void* d_out, int out_size, void* d_ws, size_t ws_size,
                              hipStream_t stream) {
    const float* x    = (const float*)d_in[0];
    const float* W1   = (const float*)d_in[1];
    const float* grid = (const float*)d_in[2];
    const float* coef = (const float*)d_in[3];
    const float* sb   = (const float*)d_in[4];
    const float* sp   = (const float*)d_in[5];
    const float* mask = (const float*)d_in[6];
    // d_in[7] = edge_index: unused by the reference computation.

    const int N = in_sizes[0] / IN_F;
    float* ws = (float*)d_ws;   // needs (SZ+SX)*256*4 = 327,680 bytes

    const int prepN = (KZ + IN_F) * OUT_F;
    kan_prep<<<(prepN + 255) / 256, 256, 0, stream>>>(W1, coef, sb, sp, mask, ws);
    kan_fused<<<(N + 15) / 16, 128, 0, stream>>>(x, grid, ws, (float*)d_out, N);
}